// GMMLoss_37477884625578
// MI455X (gfx1250) — compile-verified
//
#include <hip/hip_runtime.h>
#include <math.h>

// GMM energy loss: z[65536,64] f32, gamma[65536,16] f32 -> scalar f32.
#define NN 65536
#define DD 64
#define KK 16

typedef float v2f __attribute__((ext_vector_type(2)));
typedef float v8f __attribute__((ext_vector_type(8)));

// workspace layout (float offsets)
#define OFF_SUMG  0          // [16]
#define OFF_ZG    16         // [16][64]
#define OFF_S     1040       // [16][64][64]
#define OFF_AINV  66576      // [16][64][64]
#define OFF_B     132112     // [16][64]
#define OFF_C     133136     // [16]
#define OFF_COEF  133152     // [16]
#define OFF_CDIAG 133168     // [16]
#define OFF_EPART 133184     // [512]

// ---------------------------------------------------------------- kernel 1
// sum_g[k] = sum_n gamma[n,k];  zg[k,d] = sum_n gamma[n,k]*z[n,d]
__global__ void k_stats(const float* __restrict__ z, const float* __restrict__ g,
                        float* __restrict__ ws) {
  const int k = blockIdx.x;
  const int tid = threadIdx.x;
  const int j = tid >> 6, d = tid & 63;
  float acc = 0.f, sg = 0.f;
  for (int n = j; n < NN; n += 4) {
    float gv = g[n * KK + k];            // broadcast within 64-thread group
    acc += gv * z[n * DD + d];           // coalesced over d
    if (d == 0) sg += gv;
  }
  __shared__ float red[4][64];
  __shared__ float sgl[4];
  red[j][d] = acc;
  if (d == 0) sgl[j] = sg;
  __syncthreads();
  if (j == 0) ws[OFF_ZG + k * 64 + d] = red[0][d] + red[1][d] + red[2][d] + red[3][d];
  if (tid == 0) ws[OFF_SUMG + k] = sgl[0] + sgl[1] + sgl[2] + sgl[3];
}

// ---------------------------------------------------------------- kernel 2
// S[k] = Z^T diag(gamma_k) Z  (64x64, reduce over N) via v_wmma_f32_16x16x4_f32.
// grid = (k=16, tile=16), tile -> 16x16 output block (td,te). 8 waves split N.
__global__ void k_scatter(const float* __restrict__ z, const float* __restrict__ g,
                          float* __restrict__ ws) {
  const int k = blockIdx.x;
  const int tile = blockIdx.y;
  const int dbase = (tile >> 2) * 16, ebase = (tile & 3) * 16;
  const int tid = threadIdx.x;
  const int wave = tid >> 5, lane = tid & 31;
  const int half = lane >> 4, m = lane & 15;
  const int per = NN / 8;
  const int nb = wave * per, ne2 = nb + per;

  v8f acc = {0.f, 0.f, 0.f, 0.f, 0.f, 0.f, 0.f, 0.f};
  for (int n0 = nb; n0 < ne2; n0 += 4) {
    const int r0 = n0 + 2 * half;
    const float g0 = g[r0 * KK + k];
    const float g1 = g[(r0 + 1) * KK + k];
    const float* z0 = z + r0 * DD;
    const float* z1 = z0 + DD;
    // A (16x4, M x Kdim): A[m,j] = gamma[n0+j]*z[n0+j, dbase+m]
    // B (4x16, Kdim x N): B[j,n] = z[n0+j, ebase+n]
    v2f a, b;
    a.x = g0 * z0[dbase + m];
    a.y = g1 * z1[dbase + m];
    b.x = z0[ebase + m];
    b.y = z1[ebase + m];
    acc = __builtin_amdgcn_wmma_f32_16x16x4_f32(false, a, false, b, (short)0, acc,
                                                false, false);
  }
  // cross-wave reduction of the 16x16 accumulator tile
  __shared__ float lred[8][256];
#pragma unroll
  for (int r = 0; r < 8; ++r) lred[wave][r * 32 + lane] = acc[r];
  __syncthreads();
  float s = 0.f;
#pragma unroll
  for (int w = 0; w < 8; ++w) s += lred[w][tid];
  const int rr = tid >> 5, ln = tid & 31;
  const int drow = dbase + rr + 8 * (ln >> 4);   // D-matrix layout
  const int ecol = ebase + (ln & 15);
  ws[OFF_S + k * 4096 + drow * 64 + ecol] = s;
}

// ---------------------------------------------------------------- kernel 3
// cov = S/sum_g - mu mu^T + eps*I; Gauss-Jordan inverse w/ partial pivoting,
// det from pivots; b = inv*mu, c = mu^T inv mu, coef = phi/sqrt(det+eps),
// cov_diag partial. One workgroup per k.
__global__ void k_finalize(float* __restrict__ ws) {
  const int k = blockIdx.x;
  const int tid = threadIdx.x;
  __shared__ float Am[64][65];
  __shared__ float Iv[64][65];
  __shared__ float mu[64];
  __shared__ float fv[64];
  __shared__ float sdet;
  __shared__ int ipiv;

  const float sg = ws[OFF_SUMG + k];
  if (tid < 64) mu[tid] = ws[OFF_ZG + k * 64 + tid] / sg;
  if (tid == 0) sdet = 1.f;
  __syncthreads();
  for (int f = tid; f < 4096; f += 256) {
    const int d = f >> 6, e = f & 63;
    float c = ws[OFF_S + k * 4096 + f] / sg - mu[d] * mu[e];
    if (d == e) c += 1e-9f;
    Am[d][e] = c;
    Iv[d][e] = (d == e) ? 1.f : 0.f;
  }
  __syncthreads();
  if (tid == 0) {
    float cd = 0.f;
    for (int d = 0; d < 64; ++d) cd += 1.f / (Am[d][d] + 1e-9f);
    ws[OFF_CDIAG + k] = cd;
  }
  for (int col = 0; col < 64; ++col) {
    if (tid == 0) {
      int p = col;
      float best = fabsf(Am[col][col]);
      for (int r = col + 1; r < 64; ++r) {
        const float v = fabsf(Am[r][col]);
        if (v > best) { best = v; p = r; }
      }
      ipiv = p;
    }
    __syncthreads();
    const int p = ipiv;
    if (p != col && tid < 64) {
      float t = Am[col][tid]; Am[col][tid] = Am[p][tid]; Am[p][tid] = t;
      t = Iv[col][tid]; Iv[col][tid] = Iv[p][tid]; Iv[p][tid] = t;
    }
    __syncthreads();
    const float piv = Am[col][col];
    const float ip = 1.f / piv;
    __syncthreads();                      // all read piv before row scaling
    if (tid == 0) sdet *= (p != col) ? -piv : piv;
    if (tid < 64) { Am[col][tid] *= ip; Iv[col][tid] *= ip; }
    __syncthreads();
    if (tid < 64) fv[tid] = Am[tid][col]; // snapshot elimination factors
    __syncthreads();
    for (int f = tid; f < 4096; f += 256) {
      const int r = f >> 6, c = f & 63;
      if (r != col) {
        const float fr = fv[r];
        Am[r][c] -= fr * Am[col][c];
        Iv[r][c] -= fr * Iv[col][c];
      }
    }
    __syncthreads();
  }
  if (tid < 64) {
    float bb = 0.f;
    for (int e = 0; e < 64; ++e) bb += Iv[tid][e] * mu[e];
    fv[tid] = bb;
    ws[OFF_B + k * 64 + tid] = bb;
  }
  __syncthreads();
  if (tid == 0) {
    float c = 0.f;
    for (int d = 0; d < 64; ++d) c += mu[d] * fv[d];
    ws[OFF_C + k] = c;
    const float phi = sg / (float)NN;
    ws[OFF_COEF + k] = phi / sqrtf(sdet + 1e-9f);
  }
  for (int f = tid; f < 4096; f += 256) ws[OFF_AINV + k * 4096 + f] = Iv[f >> 6][f & 63];
}

// ---------------------------------------------------------------- kernel 4
// Per 16-sample tile: quad[n,k] = z^T A_k z - 2 b_k.z + c_k via WMMA P = Z*A_k
// plus in-register diag dot; mix accumulation over k; per-block energy partial.
__global__ void k_energy(const float* __restrict__ z, float* __restrict__ ws) {
  const int b = blockIdx.x;                 // 512 blocks * 8 tiles = 4096 tiles
  const int tid = threadIdx.x;
  const int w = tid >> 5, lane = tid & 31, half = lane >> 4, mm = lane & 15;
  __shared__ float zt[8][16][68];           // pad 68: conflict-free column reads
  __shared__ float ak[64][64];
  __shared__ float bk[64];
  __shared__ float ew[8];

  for (int f = tid; f < 8192; f += 256) {   // stage this block's 8 Z tiles
    const int t8 = f >> 10, row = (f >> 6) & 15, col = f & 63;
    zt[t8][row][col] = z[((b * 8 + t8) * 16 + row) * DD + col];
  }
  float mix[8];
#pragma unroll
  for (int r = 0; r < 8; ++r) mix[r] = 0.f;

  for (int k = 0; k < KK; ++k) {
    __syncthreads();
    for (int f = tid; f < 4096; f += 256) ak[f >> 6][f & 63] = ws[OFF_AINV + k * 4096 + f];
    if (tid < 64) bk[tid] = ws[OFF_B + k * 64 + tid];
    __syncthreads();
    const float ck = ws[OFF_C + k];
    const float coef = ws[OFF_COEF + k];
    float srow[8], sb[8];
#pragma unroll
    for (int r = 0; r < 8; ++r) { srow[r] = 0.f; sb[r] = 0.f; }
    for (int t = 0; t < 4; ++t) {           // output column tiles of P
      v8f acc = {0.f, 0.f, 0.f, 0.f, 0.f, 0.f, 0.f, 0.f};
#pragma unroll
      for (int c = 0; c < 16; ++c) {        // K chunks of 4
        v2f a, bb;
        a.x = zt[w][mm][4 * c + 2 * half];
        a.y = zt[w][mm][4 * c + 2 * half + 1];
        bb.x = ak[4 * c + 2 * half][16 * t + mm];
        bb.y = ak[4 * c + 2 * half + 1][16 * t + mm];
        acc = __builtin_amdgcn_wmma_f32_16x16x4_f32(false, a, false, bb, (short)0,
                                                    acc, false, false);
      }
      const float bval = bk[16 * t + mm];
#pragma unroll
      for (int r = 0; r < 8; ++r) {         // row m = r + 8*half on this half
        const float zc = zt[w][r + 8 * half][16 * t + mm];
        srow[r] += acc[r] * zc;             // z^T A z partial
        sb[r] += zc * bval;                 // b.z partial
      }
    }
#pragma unroll
    for (int r = 0; r < 8; ++r) {
      float s = srow[r], q = sb[r];
      for (int msk = 1; msk <= 8; msk <<= 1) {   // reduce within each 16-lane half
        s += __shfl_xor(s, msk, 32);
        q += __shfl_xor(q, msk, 32);
      }
      const float quad = s - 2.f * q + ck;
      mix[r] += coef * expf(-0.5f * quad);
    }
  }
  float e = 0.f;
#pragma unroll
  for (int r = 0; r < 8; ++r) e += -logf(mix[r] + 1e-9f);
  e += __shfl_xor(e, 16, 32);               // rows 0-7 half + rows 8-15 half
  if (lane == 0) ew[w] = e;
  __syncthreads();
  if (tid == 0) {
    float s = 0.f;
    for (int ww = 0; ww < 8; ++ww) s += ew[ww];
    ws[OFF_EPART + b] = s;
  }
}

// ---------------------------------------------------------------- kernel 5
__global__ void k_final(const float* __restrict__ ws, float* __restrict__ out) {
  if (threadIdx.x == 0 && blockIdx.x == 0) {
    float es = 0.f;
    for (int i = 0; i < 512; ++i) es += ws[OFF_EPART + i];
    float cd = 0.f;
    for (int i = 0; i < 16; ++i) cd += ws[OFF_CDIAG + i];
    out[0] = 0.1f * (es / (float)NN) + 0.005f * cd;
  }
}

extern "C" void kernel_launch(void* const* d_in, const int* in_sizes, int n_in,
                              void* d_out, int out_size, void* d_ws, size_t ws_size,
                              hipStream_t stream) {
  (void)in_sizes; (void)n_in; (void)out_size; (void)ws_size;
  const float* z = (const float*)d_in[0];
  const float* g = (const float*)d_in[1];
  float* ws = (float*)d_ws;
  float* out = (float*)d_out;

  k_stats<<<16, 256, 0, stream>>>(z, g, ws);
  k_scatter<<<dim3(16, 16), 256, 0, stream>>>(z, g, ws);
  k_finalize<<<16, 256, 0, stream>>>(ws);
  k_energy<<<512, 256, 0, stream>>>(z, ws);
  k_final<<<1, 32, 0, stream>>>(ws, out);
}